// ToRGB_11347303596089
// MI455X (gfx1250) — compile-verified
//
#include <hip/hip_runtime.h>

typedef __attribute__((ext_vector_type(2))) float v2f;
typedef __attribute__((ext_vector_type(8))) float v8f;

#define B_    4
#define CIN   512
#define COUT  3
#define S_    256
#define HW    (S_ * S_)      // 65536
#define HS    128            // skip spatial size

// ---------------------------------------------------------------------------
// Kernel A: mT[b][i][o16] = weight[o][i] * style[b][i] for o<3, zero for o=3..15
// Layout transposed (i-major, o minor) so the WMMA A-fragment load per lane is
// contiguous across the 16 M-rows: a0 = mT[(b*512+c)*16 + ln], a1 = +16 floats.
// ---------------------------------------------------------------------------
__global__ void modulate_kernel(const float* __restrict__ weight,
                                const float* __restrict__ style,
                                float* __restrict__ mT) {
  const int idx = blockIdx.x * blockDim.x + threadIdx.x;   // 0 .. B*CIN-1
  const int b = idx >> 9;
  const int i = idx & (CIN - 1);
  const float s = style[b * CIN + i];
  float4 v0;
  v0.x = weight[0 * CIN + i] * s;
  v0.y = weight[1 * CIN + i] * s;
  v0.z = weight[2 * CIN + i] * s;
  v0.w = 0.f;
  const float4 z = make_float4(0.f, 0.f, 0.f, 0.f);
  float4* p = (float4*)(mT + (size_t)idx * 16);
  p[0] = v0; p[1] = z; p[2] = z; p[3] = z;
}

// ---------------------------------------------------------------------------
// Main fused kernel: modulated 1x1 conv via V_WMMA_F32_16X16X4_F32 over K=512,
// + bias + upfirdn2d(up=2, 4x4 blur) skip upsample, all in one pass over x.
// Each wave: 4 tiles x 16 pixels; block (8 waves) covers 512 consecutive pixels.
// ---------------------------------------------------------------------------
__global__ void __launch_bounds__(256)
torgb_kernel(const float* __restrict__ x,
             const float* __restrict__ skip,
             const float* __restrict__ bias,
             const float* __restrict__ k2,     // 4x4 blur kernel (pre-scaled)
             const float* __restrict__ mT,
             float* __restrict__ out) {
  const int lane = threadIdx.x & 31;
  const int wave = threadIdx.x >> 5;
  const int hi   = lane >> 4;        // 0: K=0,1 half | 1: K=2,3 half
  const int ln   = lane & 15;        // M row (A) / N pixel (B,D)
  const int b    = blockIdx.y;
  const int pixbase = blockIdx.x * 512 + wave * 64;

  const float* xb = x  + (((size_t)b * CIN) << 16) + pixbase + ln;
  const float* mb = mT + (((size_t)b * CIN) << 4) + ln;

  v8f acc[4];
  #pragma unroll
  for (int t = 0; t < 4; ++t) acc[t] = (v8f){};

  for (int k = 0; k < CIN; k += 4) {
    const int c = k + 2 * hi;
    v2f a;
    a[0] = mb[(size_t)c * 16];         // K = c,   row ln   (L2-resident)
    a[1] = mb[(size_t)(c + 1) * 16];   // K = c+1, row ln
    const float* xp0 = xb + ((size_t)c << 16);
    const float* xp1 = xb + (((size_t)c + 1) << 16);
    #pragma unroll
    for (int t = 0; t < 4; ++t) {
      v2f bv;
      bv[0] = __builtin_nontemporal_load(xp0 + t * 16);  // K=c,   N=ln (stream)
      bv[1] = __builtin_nontemporal_load(xp1 + t * 16);  // K=c+1, N=ln
      acc[t] = __builtin_amdgcn_wmma_f32_16x16x4_f32(
          false, a, false, bv, (short)0, acc[t], false, false);
    }
  }

  // Epilogue: D rows 0..2 live in acc[t][0..2] of lanes 0..15 (hi==0).
  if (hi == 0) {
    const float bs0 = bias[0], bs1 = bias[1], bs2 = bias[2];
    float bs[3] = { bs0, bs1, bs2 };
    #pragma unroll
    for (int t = 0; t < 4; ++t) {
      const int pix = pixbase + t * 16 + ln;
      const int yy = pix >> 8;
      const int xx = pix & (S_ - 1);
      int ty0, ty1, iy0, iy1, tx0, tx1, ix0, ix1;
      if ((yy & 1) == 0) { ty0 = 0; ty1 = 2; iy0 = (yy >> 1) - 1; iy1 = yy >> 1; }
      else               { ty0 = 1; ty1 = 3; iy0 = yy >> 1;       iy1 = iy0 + 1; }
      if ((xx & 1) == 0) { tx0 = 0; tx1 = 2; ix0 = (xx >> 1) - 1; ix1 = xx >> 1; }
      else               { tx0 = 1; tx1 = 3; ix0 = xx >> 1;       ix1 = ix0 + 1; }
      const float w00 = k2[ty0 * 4 + tx0], w01 = k2[ty0 * 4 + tx1];
      const float w10 = k2[ty1 * 4 + tx0], w11 = k2[ty1 * 4 + tx1];
      const bool vy0 = (iy0 >= 0) && (iy0 < HS);
      const bool vy1 = (iy1 >= 0) && (iy1 < HS);
      const bool vx0 = (ix0 >= 0) && (ix0 < HS);
      const bool vx1 = (ix1 >= 0) && (ix1 < HS);
      #pragma unroll
      for (int o = 0; o < COUT; ++o) {
        const float* sp = skip + (((size_t)(b * COUT + o)) << 14);
        float s = 0.f;
        if (vy0 && vx0) s += w00 * sp[(iy0 << 7) + ix0];
        if (vy0 && vx1) s += w01 * sp[(iy0 << 7) + ix1];
        if (vy1 && vx0) s += w10 * sp[(iy1 << 7) + ix0];
        if (vy1 && vx1) s += w11 * sp[(iy1 << 7) + ix1];
        out[(((size_t)(b * COUT + o)) << 16) + pix] = acc[t][o] + bs[o] + s;
      }
    }
  }
}

// ---------------------------------------------------------------------------
extern "C" void kernel_launch(void* const* d_in, const int* in_sizes, int n_in,
                              void* d_out, int out_size, void* d_ws, size_t ws_size,
                              hipStream_t stream) {
  const float* x      = (const float*)d_in[0];
  const float* style  = (const float*)d_in[1];
  const float* skip   = (const float*)d_in[2];
  const float* weight = (const float*)d_in[3];
  const float* bias   = (const float*)d_in[4];
  const float* k2     = (const float*)d_in[5];
  float* out = (float*)d_out;
  float* mT  = (float*)d_ws;   // B*CIN*16 floats = 128 KB scratch

  modulate_kernel<<<dim3((B_ * CIN) / 256), dim3(256), 0, stream>>>(weight, style, mT);
  torgb_kernel<<<dim3(HW / 512, B_), dim3(256), 0, stream>>>(x, skip, bias, k2, mT, out);
}